// DiscreteFourierEncoding_7670811591196
// MI455X (gfx1250) — compile-verified
//
#include <hip/hip_runtime.h>
#include <hip/hip_bf16.h>

typedef __attribute__((ext_vector_type(2))) float v2f;
typedef __attribute__((ext_vector_type(8))) float v8f;

#define KPOW 8
#define COLS 129        // hidden+1
#define COLS_PAD 132    // padded to multiple of 4 (zero weights / zero basis in pad)
#define INV2PI 0.15915494309189535f

// v_sin_f32 takes its argument in revolutions (radians / 2pi).
__device__ __forceinline__ float sin_rev(float r) {
#if __has_builtin(__builtin_amdgcn_sinf)
    return __builtin_amdgcn_sinf(r);
#else
    return __sinf(r * 6.283185307179586f);
#endif
}

__global__ __launch_bounds__(256) void dfe_wmma_kernel(
    const float* __restrict__ e,      // [N]
    const float* __restrict__ W,      // [8][129]
    const float* __restrict__ b,      // [8]
    const float* __restrict__ scale,  // [1]
    const float* __restrict__ freq,   // [1]
    float* __restrict__ out,          // [8][N]
    int N)
{
    // Per-column LUT: basis(c) = sin(2pi * (t' * h[c] + p[c])), t' = sf*e^i/2pi
    __shared__ float2 hp[COLS_PAD];
    __shared__ float  Wlds[KPOW * COLS_PAD];

    for (int c = threadIdx.x; c < COLS_PAD; c += blockDim.x) {
        float h, p;
        if (c <= 1)        { h = 0.0f;             p = 0.25f; } // sin(pi/2)=1
        else if (c <= 65)  { h = (float)(c - 1);   p = 0.0f;  } // sin block
        else if (c <= 128) { h = (float)(c - 65);  p = 0.25f; } // cos block
        else               { h = 0.0f;             p = 0.0f;  } // pad -> 0
        hp[c] = make_float2(h, p);
    }
    for (int idx = threadIdx.x; idx < KPOW * COLS_PAD; idx += blockDim.x) {
        const int i = idx / COLS_PAD;
        const int c = idx % COLS_PAD;
        Wlds[idx] = (c < COLS) ? W[i * COLS + c] : 0.0f;
    }
    __syncthreads();

    const int lane = threadIdx.x & 31;
    const int tile = blockIdx.x * (blockDim.x >> 5) + (threadIdx.x >> 5);
    const int n_base = tile * 16;
    if (n_base >= N) return;          // wave-uniform: EXEC all-ones at WMMA

    // A-matrix 16x4 f32 layout (ISA 7.12.2): lanes 0-15 hold M=lane, K=k0,k0+1;
    // lanes 16-31 hold M=lane-16, K=k0+2,k0+3.
    const int   m    = lane & 15;
    const int   kadd = (lane >> 4) << 1;        // 0 or 2
    const float sel  = (m == 0) ? 1.0f : 0.0f;  // B column-0 selector

    const int   nrow = n_base + m;
    const float e_m  = (nrow < N) ? e[nrow] : 0.0f;
    const float sfn  = scale[0] * freq[0] * INV2PI;

    float pw = 1.0f;
    #pragma unroll 1
    for (int i = 0; i < KPOW; ++i) {
        pw *= e_m;                   // e^(i+1)
        const float t = sfn * pw;    // revolutions per harmonic step
        const float* wrow = &Wlds[i * COLS_PAD];

        v8f acc = {0.f, 0.f, 0.f, 0.f, 0.f, 0.f, 0.f, 0.f};
        #pragma unroll 1
        for (int k0 = 0; k0 < COLS_PAD; k0 += 4) {
            const int c0 = k0 + kadd;
            const float2 h0 = hp[c0];
            const float2 h1 = hp[c0 + 1];
            v2f a, bv;
            a.x = sin_rev(__builtin_fmaf(t, h0.x, h0.y));
            a.y = sin_rev(__builtin_fmaf(t, h1.x, h1.y));
            // B 4x16 f32: VGPR0 K=2*(lane>>4), VGPR1 K=2*(lane>>4)+1, col=lane&15;
            // only column 0 carries W[i].
            bv.x = sel * wrow[c0];
            bv.y = sel * wrow[c0 + 1];
            acc = __builtin_amdgcn_wmma_f32_16x16x4_f32(
                false, a, false, bv, (short)0, acc, false, false);
        }

        // C/D 16x16 f32: column 0 lives in lane 0 (M=0..7 across the 8 VGPRs)
        // and lane 16 (M=8..15).
        const float bb = b[i];
        if (m == 0) {
            const int rbase = n_base + ((lane >> 4) << 3);
            if (rbase + 8 <= N) {
                float4* o4 = (float4*)(out + (size_t)i * (size_t)N + rbase);
                o4[0] = make_float4(acc[0] + bb, acc[1] + bb, acc[2] + bb, acc[3] + bb);
                o4[1] = make_float4(acc[4] + bb, acc[5] + bb, acc[6] + bb, acc[7] + bb);
            } else {
                for (int j = 0; j < 8; ++j)
                    if (rbase + j < N) out[(size_t)i * (size_t)N + rbase + j] = acc[j] + bb;
            }
        }
    }
}

extern "C" void kernel_launch(void* const* d_in, const int* in_sizes, int n_in,
                              void* d_out, int out_size, void* d_ws, size_t ws_size,
                              hipStream_t stream) {
    const float* e  = (const float*)d_in[0];
    const float* W  = (const float*)d_in[1];  // [8,1,129] flat
    const float* b  = (const float*)d_in[2];  // [8,1] flat
    const float* sc = (const float*)d_in[3];
    const float* fr = (const float*)d_in[4];
    float* out = (float*)d_out;               // [8, N] flat

    const int N = in_sizes[0];
    const int tiles = (N + 15) / 16;
    const int wavesPerBlk = 256 / 32;
    const int grid = (tiles + wavesPerBlk - 1) / wavesPerBlk;

    dfe_wmma_kernel<<<grid, 256, 0, stream>>>(e, W, b, sc, fr, out, N);
}